// MPNN_1864015807087
// MI455X (gfx1250) — compile-verified
//
#include <hip/hip_runtime.h>

typedef __attribute__((ext_vector_type(2))) float v2f;
typedef __attribute__((ext_vector_type(8))) float v8f;

constexpr int kNodes  = 50000;
constexpr int kEdges  = 800000;
constexpr int kHid    = 64;
constexpr int kLayers = 3;
constexpr int kGraphs = 64;

// ---------------------------------------------------------------- utilities
// Vectorized fill: n must be a multiple of 4 (all our buffers are).
__global__ void zero4_kernel(float4* __restrict__ p, int n4) {
    int i = blockIdx.x * blockDim.x + threadIdx.x;
    int stride = gridDim.x * blockDim.x;
    float4 z = make_float4(0.f, 0.f, 0.f, 0.f);
    for (; i < n4; i += stride) p[i] = z;
}

// s[n] = sum of edge_attr over incoming edges; deg[n] = in-degree. Once for all layers.
__global__ void sdeg_kernel(const int* __restrict__ ei, const float* __restrict__ ea,
                            float* __restrict__ s, float* __restrict__ deg) {
    int e = blockIdx.x * blockDim.x + threadIdx.x;
    if (e >= kEdges) return;
    int dst = ei[kEdges + e];
    atomicAdd(&s[dst], ea[e]);
    atomicAdd(&deg[dst], 1.0f);
}

// u[l] = edge_w[l] @ W2[l], v[l] = edge_b[l] @ W2[l]  (W2 = rows 64..127 of node_w[l])
__global__ void uv_kernel(const float* __restrict__ edge_w, const float* __restrict__ edge_b,
                          const float* __restrict__ node_w,
                          float* __restrict__ u, float* __restrict__ v) {
    int t = blockIdx.x * blockDim.x + threadIdx.x;
    if (t >= kLayers * kHid) return;
    int l = t / kHid, n = t % kHid;
    const float* W2 = node_w + (long)l * 2 * kHid * kHid + (long)kHid * kHid;
    const float* ew = edge_w + l * kHid;
    const float* eb = edge_b + l * kHid;
    float au = 0.f, av = 0.f;
    for (int k = 0; k < kHid; ++k) {
        float w = W2[k * kHid + n];
        au = fmaf(ew[k], w, au);
        av = fmaf(eb[k], w, av);
    }
    u[t] = au; v[t] = av;
}

// aggr[dst] += x[src] over all edges; thread = (edge, feature).
// Lanes of a wave cover consecutive features of (at most two) edges: the x[src]
// gathers and the L2 atomics are both 128B-contiguous per wave.
__global__ void scatter_kernel(const float* __restrict__ x, const int* __restrict__ ei,
                               float* __restrict__ aggr) {
    long idx = (long)blockIdx.x * blockDim.x + threadIdx.x;
    int e = (int)(idx >> 6);
    int h = (int)(idx & 63);
    if (e >= kEdges) return;
    int src = ei[e];
    int dst = ei[kEdges + e];
    atomicAdd(&aggr[(long)dst * kHid + h], x[(long)src * kHid + h]);
}

// x_out = relu(aggr @ W1 + s*u + deg*v + b) via V_WMMA_F32_16X16X4_F32.
// One wave computes a 16-row x 64-col output tile (4 v8f accumulators).
__global__ __launch_bounds__(256) void update_kernel(
        const float* __restrict__ aggr, const float* __restrict__ node_w,
        const float* __restrict__ node_b, const float* __restrict__ s,
        const float* __restrict__ deg, const float* __restrict__ u,
        const float* __restrict__ v, float* __restrict__ xout, int layer) {
    int wave = threadIdx.x >> 5;
    int lane = threadIdx.x & 31;
    int gw   = blockIdx.x * (blockDim.x >> 5) + wave;
    int m0   = gw * 16;
    if (m0 >= kNodes) return;            // wave-uniform: EXEC stays all-ones for WMMA
    int half = lane >> 4;                // 0: lanes 0-15, 1: lanes 16-31
    int l15  = lane & 15;

    const float* W1 = node_w + (long)layer * 2 * kHid * kHid;   // rows 0..63

    v8f acc0 = {}, acc1 = {}, acc2 = {}, acc3 = {};
    for (int k0 = 0; k0 < kHid; k0 += 4) {
        int ka = k0 + half * 2;
        // A fragment: row M = m0+l15, K = ka, ka+1
        const float* arow = aggr + (long)(m0 + l15) * kHid + ka;
        v2f a; a.x = arow[0]; a.y = arow[1];
        // B fragments: col N = n0+l15, rows K = ka, ka+1 (4 n-tiles)
        const float* b0p = W1 + (long)ka * kHid + l15;
        const float* b1p = b0p + kHid;
        v2f b0, b1, b2, b3;
        b0.x = b0p[0];  b0.y = b1p[0];
        b1.x = b0p[16]; b1.y = b1p[16];
        b2.x = b0p[32]; b2.y = b1p[32];
        b3.x = b0p[48]; b3.y = b1p[48];
        acc0 = __builtin_amdgcn_wmma_f32_16x16x4_f32(false, a, false, b0, (short)0, acc0, false, false);
        acc1 = __builtin_amdgcn_wmma_f32_16x16x4_f32(false, a, false, b1, (short)0, acc1, false, false);
        acc2 = __builtin_amdgcn_wmma_f32_16x16x4_f32(false, a, false, b2, (short)0, acc2, false, false);
        acc3 = __builtin_amdgcn_wmma_f32_16x16x4_f32(false, a, false, b3, (short)0, acc3, false, false);
    }

    const float* U = u + layer * kHid;
    const float* V = v + layer * kHid;
    const float* B = node_b + layer * kHid;
    float un0 = U[l15], un1 = U[16 + l15], un2 = U[32 + l15], un3 = U[48 + l15];
    float vn0 = V[l15], vn1 = V[16 + l15], vn2 = V[32 + l15], vn3 = V[48 + l15];
    float bn0 = B[l15], bn1 = B[16 + l15], bn2 = B[32 + l15], bn3 = B[48 + l15];

    for (int j = 0; j < 8; ++j) {
        int m = m0 + j + half * 8;       // C/D layout: VGPR j -> row j (lanes<16) / j+8
        float sm = s[m], dm = deg[m];
        float r0 = fmaxf(acc0[j] + sm * un0 + dm * vn0 + bn0, 0.f);
        float r1 = fmaxf(acc1[j] + sm * un1 + dm * vn1 + bn1, 0.f);
        float r2 = fmaxf(acc2[j] + sm * un2 + dm * vn2 + bn2, 0.f);
        float r3 = fmaxf(acc3[j] + sm * un3 + dm * vn3 + bn3, 0.f);
        float* orow = xout + (long)m * kHid + l15;
        orow[0] = r0; orow[16] = r1; orow[32] = r2; orow[48] = r3;
    }
}

// global mean pool scatter
__global__ void pool_kernel(const float* __restrict__ x, const int* __restrict__ batch,
                            float* __restrict__ sums, float* __restrict__ counts) {
    long idx = (long)blockIdx.x * blockDim.x + threadIdx.x;
    int n = (int)(idx >> 6), h = (int)(idx & 63);
    if (n >= kNodes) return;
    int g = batch[n];
    atomicAdd(&sums[(long)g * kHid + h], x[(long)n * kHid + h]);
    if (h == 0) atomicAdd(&counts[g], 1.0f);
}

// out[g] = mean-pooled features . fc_w + fc_b
__global__ void fc_kernel(const float* __restrict__ sums, const float* __restrict__ counts,
                          const float* __restrict__ fc_w, const float* __restrict__ fc_b,
                          float* __restrict__ out) {
    int g = threadIdx.x;
    if (g >= kGraphs) return;
    float c = fmaxf(counts[g], 1.0f);
    float acc = 0.f;
    for (int h = 0; h < kHid; ++h) acc = fmaf(sums[g * kHid + h] / c, fc_w[h], acc);
    out[g] = acc + fc_b[0];
}

extern "C" void kernel_launch(void* const* d_in, const int* in_sizes, int n_in,
                              void* d_out, int out_size, void* d_ws, size_t ws_size,
                              hipStream_t stream) {
    const float* x          = (const float*)d_in[0];
    const float* edge_attr  = (const float*)d_in[1];
    const float* edge_w     = (const float*)d_in[2];
    const float* edge_b     = (const float*)d_in[3];
    const float* node_w     = (const float*)d_in[4];
    const float* node_b     = (const float*)d_in[5];
    const float* fc_w       = (const float*)d_in[6];
    const float* fc_b       = (const float*)d_in[7];
    const int*   edge_index = (const int*)d_in[8];
    const int*   batch      = (const int*)d_in[9];
    (void)in_sizes; (void)n_in; (void)out_size; (void)ws_size;

    char* ws = (char*)d_ws;
    size_t off = 0;
    auto carve = [&](size_t bytes) -> void* {
        void* p = ws + off;
        off = (off + bytes + 255) & ~(size_t)255;
        return p;
    };
    float* aggr = (float*)carve((size_t)kNodes * kHid * sizeof(float));
    float* xb0  = (float*)carve((size_t)kNodes * kHid * sizeof(float));
    float* xb1  = (float*)carve((size_t)kNodes * kHid * sizeof(float));
    float* sdeg = (float*)carve((size_t)2 * kNodes * sizeof(float));
    float* sN   = sdeg;
    float* degN = sdeg + kNodes;
    float* u    = (float*)carve((size_t)kLayers * kHid * sizeof(float));
    float* v    = (float*)carve((size_t)kLayers * kHid * sizeof(float));
    float* pool = (float*)carve((size_t)(kGraphs * kHid + kGraphs) * sizeof(float));
    float* sums   = pool;
    float* counts = pool + kGraphs * kHid;

    // one-time per-call precompute (edge statistics + fused edge-MLP weights)
    zero4_kernel<<<256, 256, 0, stream>>>((float4*)sdeg, (2 * kNodes) / 4);
    sdeg_kernel<<<(kEdges + 255) / 256, 256, 0, stream>>>(edge_index, edge_attr, sN, degN);
    uv_kernel<<<1, 256, 0, stream>>>(edge_w, edge_b, node_w, u, v);

    const float* xin = x;
    float* xout = xb0;
    for (int l = 0; l < kLayers; ++l) {
        zero4_kernel<<<1024, 256, 0, stream>>>((float4*)aggr, (kNodes * kHid) / 4);
        long scatter_threads = (long)kEdges * kHid;
        scatter_kernel<<<(unsigned)((scatter_threads + 255) / 256), 256, 0, stream>>>(
            xin, edge_index, aggr);
        int waves  = (kNodes + 15) / 16;           // 3125 tiles of 16 rows
        int blocks = (waves + 7) / 8;              // 8 waves per 256-thread block
        update_kernel<<<blocks, 256, 0, stream>>>(aggr, node_w, node_b, sN, degN, u, v, xout, l);
        xin  = xout;
        xout = (xout == xb0) ? xb1 : xb0;
    }

    zero4_kernel<<<16, 256, 0, stream>>>((float4*)pool, (kGraphs * kHid + kGraphs + 3) / 4);
    long pool_threads = (long)kNodes * kHid;
    pool_kernel<<<(unsigned)((pool_threads + 255) / 256), 256, 0, stream>>>(
        xin, batch, sums, counts);
    fc_kernel<<<1, 64, 0, stream>>>(sums, counts, fc_w, fc_b, (float*)d_out);
}